// gat_63539746177230
// MI455X (gfx1250) — compile-verified
//
#include <hip/hip_runtime.h>

#define DIM 128

typedef __attribute__((ext_vector_type(2))) float v2f;
typedef __attribute__((ext_vector_type(8))) float v8f;

#ifndef HAVE_TDM
#  if defined(__has_builtin)
#    if __has_builtin(__builtin_amdgcn_tensor_load_to_lds) && \
        __has_builtin(__builtin_amdgcn_s_wait_tensorcnt)
#      define HAVE_TDM 1
#    else
#      define HAVE_TDM 0
#    endif
#  else
#    define HAVE_TDM 0
#  endif
#endif

#if HAVE_TDM
typedef __attribute__((ext_vector_type(4))) unsigned tdm_u4;
typedef __attribute__((ext_vector_type(8))) int      tdm_i8;
typedef __attribute__((ext_vector_type(4))) int      tdm_i4;
#endif

// ---------------- utility ----------------
__global__ void k_zero(unsigned* __restrict__ p, long n) {
  long i = (long)blockIdx.x * blockDim.x + threadIdx.x;
  long st = (long)gridDim.x * blockDim.x;
  for (; i < n; i += st) p[i] = 0u;
}

// ordered-uint encoding so unsigned atomicMax == float max (handles negatives)
__device__ __forceinline__ unsigned f2ord(float f) {
  unsigned u = __float_as_uint(f);
  return (u & 0x80000000u) ? ~u : (u | 0x80000000u);
}
__device__ __forceinline__ float ord2f(unsigned u) {
  return (u & 0x80000000u) ? __uint_as_float(u & 0x7fffffffu) : __uint_as_float(~u);
}

// ---------------- h = x @ W via V_WMMA_F32_16X16X4_F32 ----------------
// 256 threads = 8 wave32; each wave computes a 16x128 strip of H.
// W staged in LDS fragment-major (64KB): pair (W[k][c],W[k+1][c]) contiguous,
// so each B fragment is one aligned ds_load_b64 (no VGPR repacking).
// X tile staged via the Tensor Data Mover with LDS padding (row stride 132
// DWORDs -> A-fragment reads hit distinct banks: bank = (4m + c) % 64).
__launch_bounds__(256)
__global__ void k_gemm_wmma(const float* __restrict__ X, const float* __restrict__ W,
                            float* __restrict__ H, int nrows) {
  __shared__ float sW[DIM * DIM];   // 64 KB, swizzled fragment-major
  __shared__ float sX[128 * 132];   // 66 KB, row-major with +4 DWORD row pad
  const int tid  = threadIdx.x;
  const int wave = tid >> 5;
  const int lane = tid & 31;
  const int row0blk = blockIdx.x * 128;

  // ---- stage W swizzled: sW[(k>>1)*256 + c*2 + (k&1)] = W[k][c] ----
  // unit = (row-pair kp, 4-col group c4); 2048 units / 256 threads = 8 each
  #pragma unroll
  for (int u = 0; u < 8; ++u) {
    int unit = u * 256 + tid;
    int kp = unit >> 5;           // 0..63
    int c4 = (unit & 31) * 4;     // 0..124
    float4 a = *(const float4*)&W[(size_t)(2 * kp) * DIM + c4];
    float4 b = *(const float4*)&W[(size_t)(2 * kp + 1) * DIM + c4];
    float* o = &sW[kp * 256 + c4 * 2];
    float4 lo; lo.x = a.x; lo.y = b.x; lo.z = a.y; lo.w = b.y;
    float4 hi; hi.x = a.z; hi.y = b.z; hi.z = a.w; hi.w = b.w;
    *(float4*)&o[0] = lo;
    *(float4*)&o[4] = hi;
  }

  // ---- stage X tile (128 rows x 128 cols, padded rows) ----
#if HAVE_TDM
  if (row0blk + 128 <= nrows) {
    if (wave == 0) {
      unsigned long long ga = (unsigned long long)(size_t)(X + (size_t)row0blk * DIM);
      unsigned rows_rem = (unsigned)(nrows - row0blk);
      tdm_u4 g0;
      g0.x = 1u;                                  // count=1 (valid user D#)
      g0.y = (unsigned)(size_t)sX;                // lds_addr (bytes)
      g0.z = (unsigned)ga;                        // global_addr[31:0]
      g0.w = (unsigned)((ga >> 32) & 0x01FFFFFFu) | (2u << 30);  // addr[56:32] | type=2
      tdm_i8 g1;
      g1.s0 = (int)((2u << 16)      // data_size = 4 bytes
                  | (1u << 20)      // pad_enable
                  | (6u << 22)      // pad_interval: 2^(6+1)=128 DWORDs
                  | (3u << 25));    // pad_amount: 4 DWORDs
      g1.s1 = (int)(128u << 16);    // tensor_dim0 = 128 (bits 48..63)
      g1.s2 = (int)((rows_rem & 0xFFFFu) << 16);          // tensor_dim1 lo16
      g1.s3 = (int)((rows_rem >> 16) | (128u << 16));     // dim1 hi16 | tile_dim0=128
      g1.s4 = 128;                  // tile_dim1 = 128, tile_dim2 = 0
      g1.s5 = 128;                  // tensor_dim0_stride = 128 elements
      g1.s6 = 0;
      g1.s7 = 0;
      tdm_i4 gz; gz.x = 0; gz.y = 0; gz.z = 0; gz.w = 0;
      tdm_i8 gz8;
      gz8.s0 = 0; gz8.s1 = 0; gz8.s2 = 0; gz8.s3 = 0;
      gz8.s4 = 0; gz8.s5 = 0; gz8.s6 = 0; gz8.s7 = 0;
      __builtin_amdgcn_tensor_load_to_lds(g0, g1, gz, gz, gz8, 0);
      __builtin_amdgcn_s_wait_tensorcnt((short)0);
    }
  } else
#endif
  {
    // cooperative fallback (partial / last block): zero-pad OOB rows
    #pragma unroll
    for (int i = 0; i < 16; ++i) {
      int idx = (i * 256 + tid) * 4;
      int r = idx >> 7;
      int c = idx & 127;
      int row = row0blk + r;
      float4 v = {0.f, 0.f, 0.f, 0.f};
      if (row < nrows) v = *(const float4*)&X[(size_t)row * DIM + c];
      *(float4*)&sX[r * 132 + c] = v;
    }
  }
  __syncthreads();

  const int row0 = row0blk + wave * 16;
  if (row0 >= nrows) return;  // uniform per-wave exit (after barrier)

  const int half = lane >> 4;   // K-pair select (A) / output row group (C)
  const int mn   = lane & 15;   // A-row m / B-col n / C-col

  v8f acc[8] = {};  // 8 N-tiles of 16x16 f32 accumulators

  const float* ax = &sX[(wave * 16 + mn) * 132];
  for (int kk = 0; kk < DIM; kk += 4) {
    // A fragment: lane holds {x[m][kk+2*half], x[m][kk+2*half+1]}
    v2f a = *(const v2f*)(ax + kk + 2 * half);
    // B fragments: contiguous pair per lane in swizzled sW
    const float* wp = &sW[((kk >> 1) + half) * 256 + mn * 2];
    #pragma unroll
    for (int nn = 0; nn < 8; ++nn) {
      v2f b = *(const v2f*)(wp + nn * 32);
      acc[nn] = __builtin_amdgcn_wmma_f32_16x16x4_f32(
          /*neg_a=*/false, a, /*neg_b=*/false, b,
          /*c_mod=*/(short)0, acc[nn], /*reuse_a=*/false, /*reuse_b=*/false);
    }
  }

  // C/D layout: VGPR r, lanes 0-15 -> row r, lanes 16-31 -> row r+8
  #pragma unroll
  for (int r = 0; r < 8; ++r) {
    int row = row0 + r + 8 * half;
    if (row < nrows) {
      float* hp = H + (size_t)row * DIM + mn;
      #pragma unroll
      for (int nn = 0; nn < 8; ++nn) hp[nn * 16] = acc[nn][r];
    }
  }
}

// ---------------- per-node attention projections ----------------
// one wave32 per node: sS[n] = h[n].a_src ; sD[n] = h[n].a_dst
__global__ void k_scores(const float* __restrict__ H, const float* __restrict__ a_src,
                         const float* __restrict__ a_dst, float* __restrict__ sS,
                         float* __restrict__ sD, int n) {
  const int wave = threadIdx.x >> 5;
  const int lane = threadIdx.x & 31;
  const int node = blockIdx.x * 8 + wave;
  if (node >= n) return;
  float4 h  = *(const float4*)&H[(size_t)node * DIM + lane * 4];
  float4 as = *(const float4*)&a_src[lane * 4];
  float4 ad = *(const float4*)&a_dst[lane * 4];
  float ss = h.x * as.x + h.y * as.y + h.z * as.z + h.w * as.w;
  float sd = h.x * ad.x + h.y * ad.y + h.z * ad.z + h.w * ad.w;
  #pragma unroll
  for (int off = 16; off > 0; off >>= 1) {
    ss += __shfl_down(ss, off);
    sd += __shfl_down(sd, off);
  }
  if (lane == 0) { sS[node] = ss; sD[node] = sd; }
}

// ---------------- edge logits + segment max ----------------
__global__ void k_edge_logit(const int* __restrict__ src, const int* __restrict__ dst,
                             const float* __restrict__ sS, const float* __restrict__ sD,
                             float* __restrict__ logits, unsigned* __restrict__ mOrd, int E) {
  int e = blockIdx.x * blockDim.x + threadIdx.x;
  if (e >= E) return;
  float L = sS[src[e]] + sD[dst[e]];
  L = (L > 0.f) ? L : 0.2f * L;  // leaky_relu slope 0.2
  logits[e] = L;
  atomicMax(&mOrd[dst[e]], f2ord(L));
}

// ---------------- exp + segment sum ----------------
__global__ void k_edge_exp(const int* __restrict__ dst, const float* __restrict__ logits,
                           const float* __restrict__ ew, const unsigned* __restrict__ mOrd,
                           float* __restrict__ evals, float* __restrict__ denom, int E) {
  int e = blockIdx.x * blockDim.x + threadIdx.x;
  if (e >= E) return;
  int d = dst[e];
  float v = __expf(logits[e] - ord2f(mOrd[d])) * ew[e];
  evals[e] = v;
  atomicAdd(&denom[d], v);
}

// ---------------- weighted scatter-add aggregation ----------------
// one wave32 per edge (4 floats/lane); grid-strided with prefetch of the
// next edge's source row (global_prefetch_b8).
__global__ void k_aggregate(const float* __restrict__ H, const int* __restrict__ src,
                            const int* __restrict__ dst, const float* __restrict__ evals,
                            const float* __restrict__ denom, float* __restrict__ Out, int E) {
  const int lane = threadIdx.x & 31;
  const int wave = threadIdx.x >> 5;
  const int nw = (blockDim.x >> 5) * gridDim.x;
  const int col = lane * 4;
  for (int e = blockIdx.x * (blockDim.x >> 5) + wave; e < E; e += nw) {
    int en = e + nw;
    if (en < E) __builtin_prefetch(&H[(size_t)src[en] * DIM], 0, 1);
    int s = src[e], d = dst[e];
    float coef = evals[e] / (denom[d] + 1e-16f);
    float4 h = *(const float4*)&H[(size_t)s * DIM + col];
    float* op = &Out[(size_t)d * DIM + col];
    atomicAdd(op + 0, h.x * coef);
    atomicAdd(op + 1, h.y * coef);
    atomicAdd(op + 2, h.z * coef);
    atomicAdd(op + 3, h.w * coef);
  }
}

// ---------------- elementwise ELU ----------------
__global__ void k_elu(const float* __restrict__ in, float* __restrict__ out, long n) {
  long i = (long)blockIdx.x * blockDim.x + threadIdx.x;
  long st = (long)gridDim.x * blockDim.x;
  for (; i < n; i += st) {
    float x = in[i];
    out[i] = (x > 0.f) ? x : (__expf(x) - 1.f);
  }
}

// ---------------- final row gather ----------------
__global__ void k_gather(const float* __restrict__ nodes, const int* __restrict__ ids,
                         float* __restrict__ out, int total) {
  int i = blockIdx.x * blockDim.x + threadIdx.x;
  if (i >= total) return;
  int row = i >> 7, col = i & (DIM - 1);
  out[i] = nodes[(size_t)ids[row] * DIM + col];
}

// ---------------- host orchestration ----------------
static void run_graph(const float* x0, int N, const int* eidx, int E, const float* ew,
                      const float* W1, const float* as1, const float* ad1,
                      const float* W2, const float* as2, const float* ad2,
                      const int* ids, int B, float* out,
                      float* hbuf, float* obuf, float* xbuf,
                      float* sS, float* sD, unsigned* mOrd, float* denom,
                      float* logits, float* evals, hipStream_t stream) {
  const int* src = eidx;
  const int* dst = eidx + E;
  const float* xin = x0;
  for (int layer = 0; layer < 2; ++layer) {
    const float* W  = layer ? W2  : W1;
    const float* as = layer ? as2 : as1;
    const float* ad = layer ? ad2 : ad1;
    long nfeat = (long)N * DIM;
    k_zero<<<1024, 256, 0, stream>>>((unsigned*)obuf, nfeat);
    k_zero<<<256, 256, 0, stream>>>(mOrd, N);
    k_zero<<<256, 256, 0, stream>>>((unsigned*)denom, N);
    k_gemm_wmma<<<(N + 127) / 128, 256, 0, stream>>>(xin, W, hbuf, N);
    k_scores<<<(N + 7) / 8, 256, 0, stream>>>(hbuf, as, ad, sS, sD, N);
    k_edge_logit<<<(E + 255) / 256, 256, 0, stream>>>(src, dst, sS, sD, logits, mOrd, E);
    k_edge_exp<<<(E + 255) / 256, 256, 0, stream>>>(dst, logits, ew, mOrd, evals, denom, E);
    k_aggregate<<<4096, 256, 0, stream>>>(hbuf, src, dst, evals, denom, obuf, E);
    k_elu<<<2048, 256, 0, stream>>>(obuf, xbuf, nfeat);
    xin = xbuf;
  }
  k_gather<<<(B * DIM + 255) / 256, 256, 0, stream>>>(xbuf, ids, out, B * DIM);
}

extern "C" void kernel_launch(void* const* d_in, const int* in_sizes, int n_in,
                              void* d_out, int out_size, void* d_ws, size_t ws_size,
                              hipStream_t stream) {
  const int*   uedg     = (const int*)d_in[0];
  const int*   iedg     = (const int*)d_in[1];
  const int*   user_id  = (const int*)d_in[2];
  const int*   item_id  = (const int*)d_in[3];
  const float* uval     = (const float*)d_in[4];
  const float* ival     = (const float*)d_in[5];
  const float* umat     = (const float*)d_in[6];
  const float* imat     = (const float*)d_in[7];
  const float* W_u1     = (const float*)d_in[8];
  const float* a_src_u1 = (const float*)d_in[9];
  const float* a_dst_u1 = (const float*)d_in[10];
  const float* W_u2     = (const float*)d_in[11];
  const float* a_src_u2 = (const float*)d_in[12];
  const float* a_dst_u2 = (const float*)d_in[13];
  const float* W_i1     = (const float*)d_in[14];
  const float* a_src_i1 = (const float*)d_in[15];
  const float* a_dst_i1 = (const float*)d_in[16];
  const float* W_i2     = (const float*)d_in[17];
  const float* a_src_i2 = (const float*)d_in[18];
  const float* a_dst_i2 = (const float*)d_in[19];

  int E_U = in_sizes[0] / 2, E_I = in_sizes[1] / 2;
  int B   = in_sizes[2];
  int N_U = in_sizes[6] / DIM, N_I = in_sizes[7] / DIM;
  size_t NMAX = (size_t)(N_U > N_I ? N_U : N_I);
  size_t EMAX = (size_t)(E_U > E_I ? E_U : E_I);

  auto al = [](size_t x) { return (x + 255) & ~(size_t)255; };
  char* p = (char*)d_ws;
  float*    hbuf   = (float*)p;    p += al(NMAX * DIM * sizeof(float));
  float*    obuf   = (float*)p;    p += al(NMAX * DIM * sizeof(float));
  float*    xbuf   = (float*)p;    p += al(NMAX * DIM * sizeof(float));
  float*    sS     = (float*)p;    p += al(NMAX * sizeof(float));
  float*    sD     = (float*)p;    p += al(NMAX * sizeof(float));
  unsigned* mOrd   = (unsigned*)p; p += al(NMAX * sizeof(unsigned));
  float*    denom  = (float*)p;    p += al(NMAX * sizeof(float));
  float*    logits = (float*)p;    p += al(EMAX * sizeof(float));
  float*    evals  = (float*)p;    p += al(EMAX * sizeof(float));

  float* out = (float*)d_out;
  // user tower -> out[0 : B*128)
  run_graph(umat, N_U, uedg, E_U, uval, W_u1, a_src_u1, a_dst_u1,
            W_u2, a_src_u2, a_dst_u2, user_id, B, out,
            hbuf, obuf, xbuf, sS, sD, mOrd, denom, logits, evals, stream);
  // item tower -> out[B*128 : 2*B*128)
  run_graph(imat, N_I, iedg, E_I, ival, W_i1, a_src_i1, a_dst_i1,
            W_i2, a_src_i2, a_dst_i2, item_id, B, out + (size_t)B * DIM,
            hbuf, obuf, xbuf, sS, sD, mOrd, denom, logits, evals, stream);
}